// MambaBlock_22101901705747
// MI455X (gfx1250) — compile-verified
//
#include <hip/hip_runtime.h>
#include <hip/hip_bf16.h>
#include <math.h>

// ---------------------------------------------------------------------------
// MambaBlock, exploiting l-broadcast structure (b=64, d=256, n=16, k=4, r=16,
// l=256). All GEMMs run on the CDNA5 matrix pipe via V_WMMA_F32_16X16X4_F32
// (full fp32, 16x16 tile per wave32). Scan tail uses the geometric closed form.
// ---------------------------------------------------------------------------

typedef float v2f __attribute__((ext_vector_type(2)));
typedef float v8f __attribute__((ext_vector_type(8)));

#define B_   64
#define D_   256
#define N_   16
#define R_   16
#define L_   256

// ---- per-row RMSNorm: one block per row, 256 threads ----------------------
__global__ void rmsnorm_rows(const float* __restrict__ x,
                             const float* __restrict__ w,
                             float* __restrict__ out) {
  int row = blockIdx.x, col = threadIdx.x;
  float v = x[row * D_ + col];
  __shared__ float red[D_];
  red[col] = v * v;
  __syncthreads();
  for (int s = D_ / 2; s > 0; s >>= 1) {
    if (col < s) red[col] += red[col + s];
    __syncthreads();
  }
  float rinv = rsqrtf(red[0] * (1.0f / D_) + 1e-5f);
  out[row * D_ + col] = v * rinv * w[col];
}

// ---- generic fp32 WMMA GEMM: C[MxN] = A[MxK] @ B[NxK]^T -------------------
// One wave per 16x16 output tile. grid = (N/16, M/16), block = 32.
// fp32 WMMA lane layout (ISA 7.12.2):
//   A 16x4 : lanes 0-15 rows M=0..15 K={0,1}; lanes 16-31 same rows K={2,3}
//   B 4x16 : lanes 0-15 cols N=0..15 K={0,1}; lanes 16-31 same cols K={2,3}
//   C 16x16: VGPR v, lane = col; lanes 0-15 -> row v, lanes 16-31 -> row v+8
__global__ void gemm_wmma_nt(const float* __restrict__ A, int lda,
                             const float* __restrict__ Bm, int ldb,
                             float* __restrict__ C, int ldc, int K) {
  int n0 = blockIdx.x * 16;
  int m0 = blockIdx.y * 16;
  int lane = threadIdx.x;
  int r  = lane & 15;          // row of A-frag / col of B-frag
  int kb = (lane >> 4) << 1;   // 0 or 2: K sub-offset held by this half-wave
  const float* arow = A  + (size_t)(m0 + r) * lda;
  const float* brow = Bm + (size_t)(n0 + r) * ldb;
  v8f c = {0.f, 0.f, 0.f, 0.f, 0.f, 0.f, 0.f, 0.f};
  #pragma unroll 4
  for (int k0 = 0; k0 < K; k0 += 4) {
    v2f a, b;
    a.x = arow[k0 + kb];
    a.y = arow[k0 + kb + 1];
    b.x = brow[k0 + kb];
    b.y = brow[k0 + kb + 1];
    c = __builtin_amdgcn_wmma_f32_16x16x4_f32(
        /*neg_a=*/false, a, /*neg_b=*/false, b,
        /*c_mod=*/(short)0, c, /*reuse_a=*/false, /*reuse_b=*/false);
  }
  int mb = (lane >> 4) << 3;   // 0 or 8
  #pragma unroll
  for (int v = 0; v < 8; ++v)
    C[(size_t)(m0 + mb + v) * ldc + n0 + r] = c[v];
}

// ---- depthwise-conv edge phases + SiLU ------------------------------------
// Input constant over l => conv output has 4 distinct values:
//   t=0: w3 ; t=1: w2+w3 ; t=2: w1+w2+w3 ; t>=3: w0+w1+w2+w3   (then +bias)
__global__ void conv_silu_kernel(const float* __restrict__ xr /*128x512*/,
                                 const float* __restrict__ conv_w,
                                 const float* __restrict__ conv_b,
                                 float* __restrict__ xs1 /*4x64x256*/,
                                 float* __restrict__ xs2 /*4x64x256*/) {
  int b = blockIdx.x, dd = threadIdx.x;
  float w0 = conv_w[dd * 4 + 0], w1 = conv_w[dd * 4 + 1];
  float w2 = conv_w[dd * 4 + 2], w3 = conv_w[dd * 4 + 3];
  float S[4] = {w3, w2 + w3, w1 + w2 + w3, w0 + w1 + w2 + w3};
  float bias = conv_b[dd];
  float x1c = xr[(size_t)b * 512 + dd];
  float x2c = xr[(size_t)(B_ + b) * 512 + dd];
  #pragma unroll
  for (int p = 0; p < 4; ++p) {
    float v1 = fmaf(x1c, S[p], bias);
    float v2 = fmaf(x2c, S[p], bias);
    xs1[(size_t)(p * B_ + b) * D_ + dd] = v1 / (1.f + expf(-v1));
    xs2[(size_t)(p * B_ + b) * D_ + dd] = v2 / (1.f + expf(-v2));
  }
}

// ---- selective scan: 3 edge steps + 253-step geometric closed form --------
__global__ void scan_kernel(const float* __restrict__ xdbl /*256x48*/,
                            const float* __restrict__ dlin /*256x256*/,
                            const float* __restrict__ xs1  /*4x64x256*/,
                            const float* __restrict__ dt_b,
                            const float* __restrict__ A_log,
                            const float* __restrict__ Dv,
                            const float* __restrict__ xr   /*128x512*/,
                            float* __restrict__ y) {
  int b = blockIdx.x, dd = threadIdx.x;
  __shared__ float Bsh[4][N_];
  __shared__ float Csh[N_];
  if (dd < 64) {
    int p = dd >> 4, nn = dd & 15;
    Bsh[p][nn] = xdbl[(size_t)(p * B_ + b) * 48 + R_ + nn];
  } else if (dd < 80) {
    int nn = dd - 64;
    Csh[nn] = xdbl[(size_t)(3 * B_ + b) * 48 + R_ + N_ + nn];
  }
  __syncthreads();

  float dl[4], xs[4];
  float tb = dt_b[dd];
  #pragma unroll
  for (int p = 0; p < 4; ++p) {
    float t = dlin[(size_t)(p * B_ + b) * D_ + dd] + tb;
    dl[p] = (t > 20.f) ? t : log1pf(expf(t));       // softplus
    xs[p] = xs1[(size_t)(p * B_ + b) * D_ + dd];
  }
  float an[N_];
  #pragma unroll
  for (int n = 0; n < N_; ++n) an[n] = -expf(A_log[dd * N_ + n]);

  float h[N_];
  #pragma unroll
  for (int n = 0; n < N_; ++n) h[n] = 0.f;

  // t = 0,1,2 (conv edge phases)
  #pragma unroll
  for (int p = 0; p < 3; ++p) {
    float du = dl[p] * xs[p];
    #pragma unroll
    for (int n = 0; n < N_; ++n) {
      float dA = expf(dl[p] * an[n]);
      h[n] = fmaf(dA, h[n], du * Bsh[p][n]);
    }
  }
  // t = 3..255 : constant (a, u) => h = a^253 h + u*(1-a^253)/(1-a)
  float acc = 0.f;
  {
    float du = dl[3] * xs[3];
    #pragma unroll
    for (int n = 0; n < N_; ++n) {
      float x = dl[3] * an[n];                 // strictly < 0
      float e1 = expm1f(x);
      float eN = expm1f(253.f * x);
      float geom = (e1 != 0.f) ? (eN / e1) : 253.f;
      float pN = eN + 1.f;                     // exp(253*x)
      h[n] = fmaf(pN, h[n], du * Bsh[3][n] * geom);
      acc = fmaf(h[n], Csh[n], acc);
    }
  }
  acc = fmaf(xs[3], Dv[dd], acc);
  float r1 = xr[(size_t)b * 512 + D_ + dd];    // res1 (constant over l)
  acc *= r1 / (1.f + expf(-r1));               // * silu(res1)
  y[(size_t)b * D_ + dd] = acc;
}

// ---------------------------------------------------------------------------
extern "C" void kernel_launch(void* const* d_in, const int* in_sizes, int n_in,
                              void* d_out, int out_size, void* d_ws, size_t ws_size,
                              hipStream_t stream) {
  const float* x_1     = (const float*)d_in[0];
  const float* x_2     = (const float*)d_in[1];
  const float* norm_w  = (const float*)d_in[2];
  const float* conv_w  = (const float*)d_in[3];
  const float* conv_b  = (const float*)d_in[4];
  const float* in_w    = (const float*)d_in[5];   // (512,256)
  const float* xproj_w = (const float*)d_in[6];   // (48,256)
  const float* dt_w    = (const float*)d_in[7];   // (256,16)
  const float* dt_b    = (const float*)d_in[8];
  const float* out_w   = (const float*)d_in[9];   // (256,256)
  const float* A_log   = (const float*)d_in[10];  // (256,16)
  const float* Dv      = (const float*)d_in[11];
  (void)in_sizes; (void)n_in; (void)out_size; (void)ws_size;

  float* ws   = (float*)d_ws;
  float* XN   = ws;                   // 128x256  (rows 0-63: xn1, 64-127: xn2)
  float* XR   = XN   + 128 * 256;     // 128x512  (x1c|res1 ; x2c|res2)
  float* XS1  = XR   + 128 * 512;     // 4x64x256 silu(conv(x1c)) phases
  float* XS2  = XS1  + 4 * 64 * 256;  // 4x64x256 silu(conv(x2c)) phases
  float* XDBL = XS2  + 4 * 64 * 256;  // 256x48   [delta_r | B | C] per phase
  float* DLIN = XDBL + 256 * 48;      // 256x256  delta_r @ dt_w^T
  float* Y    = DLIN + 256 * 256;     // 64x256   y at t = l-1
  float* Z    = Y    + 64 * 256;      // 64x256   y @ out_w^T

  // 1) rmsnorm of both inputs into stacked 128x256
  rmsnorm_rows<<<64, 256, 0, stream>>>(x_1, norm_w, XN);
  rmsnorm_rows<<<64, 256, 0, stream>>>(x_2, norm_w, XN + 64 * 256);

  // 2) in-projection: (128x256) @ (512x256)^T -> 128x512
  gemm_wmma_nt<<<dim3(512 / 16, 128 / 16), 32, 0, stream>>>(
      XN, 256, in_w, 256, XR, 512, 256);

  // 3) conv edge phases + SiLU
  conv_silu_kernel<<<64, 256, 0, stream>>>(XR, conv_w, conv_b, XS1, XS2);

  // 4) x-projection: (256x256) @ (48x256)^T -> 256x48
  gemm_wmma_nt<<<dim3(48 / 16, 256 / 16), 32, 0, stream>>>(
      XS2, 256, xproj_w, 256, XDBL, 48, 256);

  // 5) dt-projection: delta_r (256x16, lda=48) @ (256x16)^T -> 256x256
  gemm_wmma_nt<<<dim3(256 / 16, 256 / 16), 32, 0, stream>>>(
      XDBL, 48, dt_w, 16, DLIN, 256, 16);

  // 6) selective scan with closed-form tail -> Y (64x256)
  scan_kernel<<<64, 256, 0, stream>>>(XDBL, DLIN, XS1, dt_b, A_log, Dv, XR, Y);

  // 7) out-projection: (64x256) @ (256x256)^T -> 64x256
  gemm_wmma_nt<<<dim3(256 / 16, 64 / 16), 32, 0, stream>>>(
      Y, 256, out_w, 256, Z, 256, 256);

  // 8) final rmsnorm -> d_out (64x256 fp32)
  rmsnorm_rows<<<64, 256, 0, stream>>>(Z, norm_w, (float*)d_out);
}